// PositionRelationEncodeUnit_67731634258425
// MI455X (gfx1250) — compile-verified
//
#include <hip/hip_runtime.h>

typedef __attribute__((ext_vector_type(2))) float v2f;
typedef __attribute__((ext_vector_type(4))) float v4f;
typedef __attribute__((ext_vector_type(8))) float v8f;

static constexpr int N_OBJ = 1024;
static constexpr int DIM   = 512;
static constexpr int DBB   = 64;
static constexpr int E_N   = 32768;
static constexpr int HID   = 1024;
static constexpr int KX    = DIM + DIM + DBB;   // 1088
static constexpr int TM    = 32;                // edges per block
static constexpr int XS    = KX + 4;            // 1092: pad -> bank-conflict-free ds_load_b64
static constexpr int HS    = HID + 4;           // 1028
static constexpr size_t LDS_WORDS = (size_t)TM * XS + (size_t)TM * HS + 64;
static constexpr size_t LDS_BYTES = LDS_WORDS * 4;   // ~271.6 KB (< 320 KB/WGP)

__device__ __forceinline__ v8f wmma_f32(v2f a, v2f b, v8f c) {
    // (neg_a, A, neg_b, B, c_mod, C, reuse_a, reuse_b)
    return __builtin_amdgcn_wmma_f32_16x16x4_f32(false, a, false, b, (short)0, c,
                                                 false, false);
}

__global__ void __launch_bounds__(256, 1)
mlp_edges(const float* __restrict__ feats,
          const float* __restrict__ bbe,
          const long long* __restrict__ pairs,
          const float* __restrict__ W1, const float* __restrict__ b1,
          const float* __restrict__ W2, const float* __restrict__ b2,
          float* __restrict__ seg, float* __restrict__ cnt)
{
    extern __shared__ float smem[];
    float* xs   = smem;                       // [TM][XS]
    float* hs   = smem + (size_t)TM * XS;     // [TM][HS]
    int*   subs = (int*)(hs + (size_t)TM * HS); // [2*TM]: sub rows then obj rows

    const int tid  = threadIdx.x;
    const int e0   = blockIdx.x * TM;
    const int lane = tid & 31;
    const int wv   = tid >> 5;        // 0..7
    const int hi   = lane >> 4;       // 0/1 (upper half-wave)
    const int ln   = lane & 15;

    if (tid < TM) {
        subs[tid]      = (int)pairs[(size_t)(e0 + tid) * 2 + 0];
        subs[TM + tid] = (int)pairs[(size_t)(e0 + tid) * 2 + 1];
    }
    __syncthreads();

    // segment counts: one atomic per edge
    if (tid < TM) atomicAdd(&cnt[subs[tid]], 1.0f);

    // ---- stage x = [f_sub | f_obj | bb] tile into LDS (float4, coalesced per row) ----
    constexpr int SEG4 = KX / 4;   // 272
    for (int idx = tid; idx < TM * SEG4; idx += 256) {
        int r  = idx / SEG4;
        int c4 = idx - r * SEG4;
        int s = subs[r], o = subs[TM + r];
        v4f v;
        if (c4 < DIM / 4) {
            v = *(const v4f*)(feats + (size_t)s * DIM + c4 * 4);
        } else if (c4 < DIM / 2) {
            v = *(const v4f*)(feats + (size_t)o * DIM + (c4 - DIM / 4) * 4);
        } else {
            v = *(const v4f*)(bbe + ((size_t)s * N_OBJ + o) * DBB + (c4 - DIM / 2) * 4);
        }
        *(v4f*)(xs + (size_t)r * XS + c4 * 4) = v;
    }
    __syncthreads();

    // ---- layer 1: h = relu(x @ W1 + b1); wave wv owns hidden cols [wv*128, wv*128+128) ----
    {
        v8f acc[2][8];
        #pragma unroll
        for (int m = 0; m < 2; ++m)
            #pragma unroll
            for (int n = 0; n < 8; ++n)
                acc[m][n] = (v8f){0.f,0.f,0.f,0.f,0.f,0.f,0.f,0.f};

        const int nbase = wv * 128;
        for (int k4 = 0; k4 < KX; k4 += 4) {
            v2f a0 = *(const v2f*)(xs + (size_t)ln * XS        + k4 + 2 * hi);
            v2f a1 = *(const v2f*)(xs + (size_t)(16 + ln) * XS + k4 + 2 * hi);
            #pragma unroll
            for (int n = 0; n < 8; ++n) {
                const float* wp = W1 + (size_t)(k4 + 2 * hi) * HID + nbase + n * 16 + ln;
                v2f b; b.x = wp[0]; b.y = wp[HID];
                acc[0][n] = wmma_f32(a0, b, acc[0][n]);
                acc[1][n] = wmma_f32(a1, b, acc[1][n]);
            }
        }
        #pragma unroll
        for (int n = 0; n < 8; ++n) {
            float bias = b1[nbase + n * 16 + ln];
            #pragma unroll
            for (int m = 0; m < 2; ++m)
                #pragma unroll
                for (int v = 0; v < 8; ++v) {
                    float val = acc[m][n][v] + bias;
                    val = val > 0.f ? val : 0.f;
                    int row = 16 * m + v + 8 * hi;
                    hs[(size_t)row * HS + nbase + n * 16 + ln] = val;
                }
        }
    }
    __syncthreads();

    // ---- layer 2: out = h @ W2 + b2; wave wv owns out cols [wv*64, wv*64+64) ----
    {
        v8f acc[2][4];
        #pragma unroll
        for (int m = 0; m < 2; ++m)
            #pragma unroll
            for (int n = 0; n < 4; ++n)
                acc[m][n] = (v8f){0.f,0.f,0.f,0.f,0.f,0.f,0.f,0.f};

        const int nbase = wv * 64;
        for (int k4 = 0; k4 < HID; k4 += 4) {
            v2f a0 = *(const v2f*)(hs + (size_t)ln * HS        + k4 + 2 * hi);
            v2f a1 = *(const v2f*)(hs + (size_t)(16 + ln) * HS + k4 + 2 * hi);
            #pragma unroll
            for (int n = 0; n < 4; ++n) {
                const float* wp = W2 + (size_t)(k4 + 2 * hi) * DIM + nbase + n * 16 + ln;
                v2f b; b.x = wp[0]; b.y = wp[DIM];
                acc[0][n] = wmma_f32(a0, b, acc[0][n]);
                acc[1][n] = wmma_f32(a1, b, acc[1][n]);
            }
        }
        #pragma unroll
        for (int n = 0; n < 4; ++n) {
            float bias = b2[nbase + n * 16 + ln];
            #pragma unroll
            for (int m = 0; m < 2; ++m)
                #pragma unroll
                for (int v = 0; v < 8; ++v) {
                    int row  = 16 * m + v + 8 * hi;       // edge row in tile
                    int srow = subs[row];                 // segment target
                    atomicAdd(&seg[(size_t)srow * DIM + nbase + n * 16 + ln],
                              acc[m][n][v] + bias);
                }
        }
    }
}

__global__ void zero_ws(float* __restrict__ p, int n) {
    int i = blockIdx.x * blockDim.x + threadIdx.x;
    if (i < n) p[i] = 0.f;
}

__global__ void finalize_feats(const float* __restrict__ feats,
                               const float* __restrict__ seg,
                               const float* __restrict__ cnt,
                               float* __restrict__ out) {
    int i = blockIdx.x * blockDim.x + threadIdx.x;
    if (i < N_OBJ * DIM) {
        int r = i / DIM;
        out[i] = (feats[i] + seg[i]) / (1.0f + cnt[r]);
    }
}

__global__ void copy_bbe(const v4f* __restrict__ src, v4f* __restrict__ dst) {
    size_t i = (size_t)blockIdx.x * blockDim.x + threadIdx.x;
    dst[i] = src[i];
}

__global__ void copy_pairs(const long long* __restrict__ pairs, float* __restrict__ dst, int n) {
    int i = blockIdx.x * blockDim.x + threadIdx.x;
    if (i < n) dst[i] = (float)pairs[i];
}

extern "C" void kernel_launch(void* const* d_in, const int* in_sizes, int n_in,
                              void* d_out, int out_size, void* d_ws, size_t ws_size,
                              hipStream_t stream) {
    const float*     feats = (const float*)d_in[0];
    const float*     bbe   = (const float*)d_in[1];
    const long long* pairs = (const long long*)d_in[2];
    const float*     W1    = (const float*)d_in[3];
    const float*     b1    = (const float*)d_in[4];
    const float*     W2    = (const float*)d_in[5];
    const float*     b2    = (const float*)d_in[6];
    float* out = (float*)d_out;

    float* seg = (float*)d_ws;            // N_OBJ * DIM accumulators
    float* cnt = seg + (size_t)N_OBJ * DIM;

    // zero segment accumulators + counts each call (deterministic)
    int zn = N_OBJ * DIM + N_OBJ;
    zero_ws<<<(zn + 255) / 256, 256, 0, stream>>>(seg, zn);

    // allow >64KB dynamic LDS (host-side attribute set; not a stream op)
    hipFuncSetAttribute((const void*)mlp_edges,
                        hipFuncAttributeMaxDynamicSharedMemorySize, (int)LDS_BYTES);

    mlp_edges<<<E_N / TM, 256, LDS_BYTES, stream>>>(feats, bbe, pairs, W1, b1, W2, b2,
                                                    seg, cnt);

    finalize_feats<<<(N_OBJ * DIM) / 256, 256, 0, stream>>>(feats, seg, cnt, out);

    // passthrough outputs: bboxes_embedding then pairs (as float)
    float* out_bbe = out + (size_t)N_OBJ * DIM;
    size_t n4 = (size_t)N_OBJ * N_OBJ * DBB / 4;   // 16,777,216 float4
    copy_bbe<<<(unsigned)(n4 / 256), 256, 0, stream>>>((const v4f*)bbe, (v4f*)out_bbe);

    float* out_pairs = out_bbe + (size_t)N_OBJ * N_OBJ * DBB;
    copy_pairs<<<(E_N * 2) / 256, 256, 0, stream>>>(pairs, out_pairs, E_N * 2);
}